// Flow_55508157334123
// MI455X (gfx1250) — compile-verified
//
#include <hip/hip_runtime.h>
#include <hip/hip_bf16.h>
#include <stdint.h>

namespace {
constexpr int kH = 512;
constexpr int kW = 512;
constexpr int kC = 3;
constexpr int kB = 32;
constexpr int kHW = kH * kW;
constexpr int kBlock = 256;
}

// Bilinear flow-warp: out[b,c,y,x] = sum_{4 corners} w * x[b,c,clamp(yc),clamp(xc)]
// Thread = one (b, pixel). Channel loop inside amortizes the weight math.
__global__ __launch_bounds__(kBlock) void flow_warp_kernel(
    const float* __restrict__ x,
    const float* __restrict__ flow,
    float* __restrict__ out)
{
    __shared__ float s_fy[kBlock];
    __shared__ float s_fx[kBlock];

    const int t = threadIdx.x;
    const int p = blockIdx.x * kBlock + t;   // pixel index in [0, H*W)
    const int b = blockIdx.y;                // batch index

    // ---- Stage this tile's flow values through LDS via the gfx1250 async
    // global->LDS path (ASYNCcnt-tracked). A generic pointer to LDS carries
    // the workgroup-relative LDS byte offset in its low 32 bits
    // which is what VDST of the async load wants.
    {
        const float* g_fy = flow + p;          // flow[0] plane: y displacement
        const float* g_fx = flow + kHW + p;    // flow[1] plane: x displacement
        unsigned l_fy = (unsigned)(uintptr_t)&s_fy[t];
        unsigned l_fx = (unsigned)(uintptr_t)&s_fx[t];
        asm volatile("global_load_async_to_lds_b32 %0, %1, off"
                     :: "v"(l_fy), "v"(g_fy) : "memory");
        asm volatile("global_load_async_to_lds_b32 %0, %1, off"
                     :: "v"(l_fx), "v"(g_fx) : "memory");
        asm volatile("s_wait_asynccnt 0" ::: "memory");
    }
    __syncthreads();

    const int py = p >> 9;          // p / 512
    const int px = p & (kW - 1);    // p % 512

    const float* __restrict__ xb = x + (size_t)b * (kC * kHW);

    // Prime the cache hierarchy for this thread's neighborhood of each channel
    // plane. flow ~ N(0, 0.1), so &plane[p] is an excellent speculative
    // prefetch address; prefetch is only a hint so correctness is unaffected.
    #pragma unroll
    for (int c = 0; c < kC; ++c)
        __builtin_prefetch(xb + (size_t)c * kHW + p, 0, 3);

    // ---- Bilinear weights (match reference: base = floor(samp),
    // wts = (1 - |corner - samp|_y) * (1 - |corner - samp|_x), computed on
    // UNCLAMPED coords; gather indices use CLAMPED coords).
    const float sy = (float)py + s_fy[t];
    const float sx = (float)px + s_fx[t];
    const float fy0 = floorf(sy);
    const float fx0 = floorf(sx);
    const int y0 = (int)fy0;
    const int x0 = (int)fx0;
    const float ay = sy - fy0;                 // in [0,1)
    const float ax = sx - fx0;
    const float w00 = (1.0f - ay) * (1.0f - ax);
    const float w01 = (1.0f - ay) * ax;
    const float w10 = ay * (1.0f - ax);
    const float w11 = ay * ax;

    const int y0c = min(max(y0,     0), kH - 1);
    const int y1c = min(max(y0 + 1, 0), kH - 1);
    const int x0c = min(max(x0,     0), kW - 1);
    const int x1c = min(max(x0 + 1, 0), kW - 1);

    const int i00 = y0c * kW + x0c;
    const int i01 = y0c * kW + x1c;
    const int i10 = y1c * kW + x0c;
    const int i11 = y1c * kW + x1c;

    float* __restrict__ ob = out + (size_t)b * (kC * kHW) + p;

    #pragma unroll
    for (int c = 0; c < kC; ++c) {
        const float* __restrict__ pl = xb + (size_t)c * kHW;
        const float v = fmaf(w00, pl[i00],
                        fmaf(w01, pl[i01],
                        fmaf(w10, pl[i10], w11 * pl[i11])));
        // Output is written once and never re-read: NT store keeps x resident
        // in the 192 MB L2 instead of polluting it with the 100 MB result.
        __builtin_nontemporal_store(v, ob + (size_t)c * kHW);
    }
}

extern "C" void kernel_launch(void* const* d_in, const int* in_sizes, int n_in,
                              void* d_out, int out_size, void* d_ws, size_t ws_size,
                              hipStream_t stream) {
    const float* x    = (const float*)d_in[0];   // [32,3,512,512] f32
    const float* flow = (const float*)d_in[1];   // [2,512,512]    f32
    float* out        = (float*)d_out;           // [32,3,512,512] f32

    dim3 grid(kHW / kBlock, kB, 1);              // (1024, 32) blocks
    flow_warp_kernel<<<grid, dim3(kBlock, 1, 1), 0, stream>>>(x, flow, out);

    (void)in_sizes; (void)n_in; (void)out_size; (void)d_ws; (void)ws_size;
}